// ProbAttention_188978561553
// MI455X (gfx1250) — compile-verified
//
#include <hip/hip_runtime.h>
#include <hip/hip_bf16.h>

// Shapes fixed by the reference: B=2, L=2048, dim=2, H=8, D=64, sample_k=u=40
#define BB   2
#define LL   2048
#define DIMM 2
#define HH   8
#define DD   64
#define SK   40
#define UU   40
#define NGRP (BB*HH*DIMM)     // 32 groups (b,h,d)
#define ROWSTRIDE (DIMM*HH*DD) // 1024 floats between consecutive l in q/k/v

typedef float v2f __attribute__((ext_vector_type(2)));
typedef float v8f __attribute__((ext_vector_type(8)));

__device__ inline void atomicMaxF(float* addr, float val) {
    int* ia = (int*)addr;
    int cur = __float_as_int(*addr);
    while (val > __int_as_float(cur)) {
        int old = atomicCAS(ia, cur, __float_as_int(val));
        if (old == cur) break;
        cur = old;
    }
}

// ---------------------------------------------------------------------------
// Kernel A: sparsity measure M[b,h,d,l] = max_s(q.k_s) - sum_s(q.k_s)/L
// one wave32 per (b,h,d,l) row
// ---------------------------------------------------------------------------
__global__ void __launch_bounds__(256)
sample_scores_kernel(const float* __restrict__ q, const float* __restrict__ k,
                     const long long* __restrict__ isamp, float* __restrict__ Mws) {
    int wave = (blockIdx.x * blockDim.x + threadIdx.x) >> 5;
    int lane = threadIdx.x & 31;
    // wave id = ((b*H + h)*dim + d)*L + l  (l fastest)
    int l   = wave & (LL - 1);
    int rem = wave >> 11;
    int d   = rem & (DIMM - 1);
    rem >>= 1;
    int h   = rem & (HH - 1);
    int b   = rem >> 3;

    const float* qrow = q + ((((long)b*LL + l)*DIMM + d)*HH + h)*DD;
    float2 qv = *(const float2*)(qrow + lane*2);

    const float* kbase = k + ((long)b*LL*ROWSTRIDE) + (long)d*HH*DD + (long)h*DD;

    float maxv = -3.0e38f, sumv = 0.f;
    for (int s = 0; s < SK; ++s) {
        long long kidx = isamp[(long)l*SK + s];
        const float* krow = kbase + kidx * ROWSTRIDE;
        float2 kv = *(const float2*)(krow + lane*2);
        float dot = qv.x*kv.x + qv.y*kv.y;
        for (int off = 16; off > 0; off >>= 1)
            dot += __shfl_xor(dot, off, 32);
        maxv = fmaxf(maxv, dot);
        sumv += dot;
    }
    if (lane == 0)
        Mws[wave] = maxv - sumv * (1.0f / (float)LL);
}

// ---------------------------------------------------------------------------
// Kernel B: top-40 indices per (b,h,d) group (iterative argmax, tie -> low idx)
// ---------------------------------------------------------------------------
__global__ void __launch_bounds__(256)
topk_kernel(const float* __restrict__ Mws, int* __restrict__ idx_ws) {
    int g = blockIdx.x;
    __shared__ float vals[LL];
    __shared__ float bmax[256];
    __shared__ int   bidx[256];
    int t = threadIdx.x;
    for (int i = t; i < LL; i += 256) vals[i] = Mws[(long)g*LL + i];
    __syncthreads();
    for (int sel = 0; sel < UU; ++sel) {
        float best = -3.0e38f; int bi = LL;
        for (int i = t; i < LL; i += 256) {
            float v = vals[i];
            if (v > best || (v == best && i < bi)) { best = v; bi = i; }
        }
        bmax[t] = best; bidx[t] = bi;
        __syncthreads();
        for (int s = 128; s > 0; s >>= 1) {
            if (t < s) {
                if (bmax[t+s] > bmax[t] || (bmax[t+s] == bmax[t] && bidx[t+s] < bidx[t])) {
                    bmax[t] = bmax[t+s]; bidx[t] = bidx[t+s];
                }
            }
            __syncthreads();
        }
        if (t == 0) { idx_ws[g*UU + sel] = bidx[0]; vals[bidx[0]] = -3.0e38f; }
        __syncthreads();
    }
}

// ---------------------------------------------------------------------------
// Kernel C: V_mean over L and broadcast-fill the whole output slab
// out layout (B, dim, H, L, D)
// ---------------------------------------------------------------------------
__global__ void __launch_bounds__(256)
vmean_fill_kernel(const float* __restrict__ v, float* __restrict__ out) {
    int g = blockIdx.x;
    int d = g & 1, h = (g >> 1) & 7, b = g >> 4;
    __shared__ float part[4][DD];
    __shared__ float vm[DD];
    int t = threadIdx.x, e = t & 63, p = t >> 6;

    const float* vbase = v + ((long)b*LL*ROWSTRIDE) + (long)d*HH*DD + (long)h*DD;
    float s = 0.f;
    for (int l = p; l < LL; l += 4)
        s += vbase[(long)l*ROWSTRIDE + e];
    part[p][e] = s;
    __syncthreads();
    if (p == 0)
        vm[e] = (part[0][e] + part[1][e] + part[2][e] + part[3][e]) * (1.0f / (float)LL);
    __syncthreads();
    float m = vm[e];
    float* obase = out + (long)b*(DIMM*HH*LL*DD) + (long)d*(HH*LL*DD) + (long)h*(LL*DD);
    for (int l = p; l < LL; l += 4)
        obase[(long)l*DD + e] = m;
}

// ---------------------------------------------------------------------------
// Kernel E: dense attention on the 40 winners, fp32 WMMA 16x16x4.
// One block per (b,h,d) group; 384 threads = 12 waves, wave -> (m, d-slice).
// ---------------------------------------------------------------------------
__global__ void __launch_bounds__(384)
topu_attention_kernel(const float* __restrict__ q, const float* __restrict__ k,
                      const float* __restrict__ v, const int* __restrict__ idx_ws,
                      float* __restrict__ out) {
    int g = blockIdx.x;
    int d = g & 1, h = (g >> 1) & 7, b = g >> 4;

    __shared__ float Qs[48 * DD];      // gathered+scaled Q_reduce (padded to 48 rows)
    __shared__ float rowmax[48];
    __shared__ float rowsum[48];
    __shared__ float Ptile[12][16 * 16]; // per-wave C-layout -> A-layout staging

    int t = threadIdx.x;
    int lane = t & 31;
    int wave = t >> 5;
    int ln   = lane & 15;       // column / row within half
    int half = lane >> 4;       // 0: lanes 0-15, 1: lanes 16-31

    const float* kbase = k + ((long)b*LL*ROWSTRIDE) + (long)d*HH*DD + (long)h*DD;
    const float* vbase = v + ((long)b*LL*ROWSTRIDE) + (long)d*HH*DD + (long)h*DD;
    const float* qbase = q + ((long)b*LL*ROWSTRIDE) + (long)d*HH*DD + (long)h*DD;

    // gather scaled Q_reduce rows (1/sqrt(64) = 0.125); pad rows >= 40 with zero
    for (int i = t; i < 48 * DD; i += 384) {
        int r = i >> 6, e = i & 63;
        float val = 0.f;
        if (r < UU) {
            int grow = idx_ws[g*UU + r];
            val = qbase[(long)grow*ROWSTRIDE + e] * 0.125f;
        }
        Qs[i] = val;
    }
    if (t < 48) { rowmax[t] = -3.0e38f; rowsum[t] = 0.f; }
    __syncthreads();

    // ------------- phase 1: row max of scores over all 2048 keys -------------
    {
        int m = wave >> 2;                  // waves 0-3 -> m0, 4-7 -> m1, 8-11 -> m2
        float lmax[8];
        for (int j = 0; j < 8; ++j) lmax[j] = -3.0e38f;
        for (int it = 0; it < 32; ++it) {
            int kt = ((wave & 3) << 5) + it; // 0..127
            v8f S = {};
            for (int kk = 0; kk < DD; kk += 4) {
                int acol = kk + half*2;
                v2f a; a.x = Qs[(m*16 + ln)*DD + acol];
                       a.y = Qs[(m*16 + ln)*DD + acol + 1];
                const float* krow = kbase + (long)(kt*16 + ln)*ROWSTRIDE + acol;
                v2f bf; bf.x = krow[0]; bf.y = krow[1];
                S = __builtin_amdgcn_wmma_f32_16x16x4_f32(
                        false, a, false, bf, (short)0, S, false, false);
            }
            for (int j = 0; j < 8; ++j) lmax[j] = fmaxf(lmax[j], S[j]);
        }
        for (int j = 0; j < 8; ++j)
            atomicMaxF(&rowmax[m*16 + half*8 + j], lmax[j]);
    }
    __syncthreads();

    // ------------- phase 2: ctx = exp(S - max) @ V, plus row sums -------------
    int m   = wave >> 2;   // 0..2 : 16-row tile of queries
    int dsl = wave & 3;    // 0..3 : 16-col slice of D
    v8f acc = {};
    float psum[8];
    for (int j = 0; j < 8; ++j) psum[j] = 0.f;
    float* Pt = &Ptile[wave][0];

    for (int kt = 0; kt < 128; ++kt) {
        v8f S = {};
        for (int kk = 0; kk < DD; kk += 4) {
            int acol = kk + half*2;
            v2f a; a.x = Qs[(m*16 + ln)*DD + acol];
                   a.y = Qs[(m*16 + ln)*DD + acol + 1];
            const float* krow = kbase + (long)(kt*16 + ln)*ROWSTRIDE + acol;
            v2f bf; bf.x = krow[0]; bf.y = krow[1];
            S = __builtin_amdgcn_wmma_f32_16x16x4_f32(
                    false, a, false, bf, (short)0, S, false, false);
        }
        // P = exp(S - rowmax); stage C-layout -> row-major tile in LDS
        for (int j = 0; j < 8; ++j) {
            float p = __expf(S[j] - rowmax[m*16 + half*8 + j]);
            psum[j] += p;
            Pt[(half*8 + j)*16 + ln] = p;
        }
        // acc += P(16x16) @ V(16x16 slice), K=16 in 4 wmma steps
        for (int kk = 0; kk < 16; kk += 4) {
            int acol = kk + half*2;
            v2f a; a.x = Pt[ln*16 + acol];
                   a.y = Pt[ln*16 + acol + 1];
            const float* vp = vbase + (long)(kt*16 + acol)*ROWSTRIDE + dsl*16 + ln;
            v2f bf; bf.x = vp[0]; bf.y = vp[ROWSTRIDE];
            acc = __builtin_amdgcn_wmma_f32_16x16x4_f32(
                    false, a, false, bf, (short)0, acc, false, false);
        }
    }
    if (dsl == 0) {
        for (int j = 0; j < 8; ++j)
            atomicAdd(&rowsum[m*16 + half*8 + j], psum[j]);
    }
    __syncthreads();

    // ------------- phase 3: normalize and scatter the 40 winner rows ---------
    float* obase = out + (long)b*(DIMM*HH*LL*DD) + (long)d*(HH*LL*DD) + (long)h*(LL*DD);
    for (int j = 0; j < 8; ++j) {
        int r = m*16 + half*8 + j;
        if (r < UU) {
            int grow = idx_ws[g*UU + r];
            obase[(long)grow*DD + dsl*16 + ln] = acc[j] / rowsum[r];
        }
    }
}

// ---------------------------------------------------------------------------
extern "C" void kernel_launch(void* const* d_in, const int* in_sizes, int n_in,
                              void* d_out, int out_size, void* d_ws, size_t ws_size,
                              hipStream_t stream) {
    const float* q = (const float*)d_in[0];
    const float* k = (const float*)d_in[1];
    const float* v = (const float*)d_in[2];
    // d_in[3] = attn_mask (unused by the reference computation)
    const long long* isamp = (const long long*)d_in[4];
    float* out = (float*)d_out;

    float* Mws    = (float*)d_ws;                         // 32*2048 floats
    int*   idx_ws = (int*)((char*)d_ws + (size_t)NGRP*LL*sizeof(float)); // 32*40 ints

    // A: sampling scores. 65536 waves -> 8192 blocks of 256 threads.
    sample_scores_kernel<<<(BB*HH*DIMM*LL)/8, 256, 0, stream>>>(q, k, isamp, Mws);
    // B: top-40 per group
    topk_kernel<<<NGRP, 256, 0, stream>>>(Mws, idx_ws);
    // C: V_mean broadcast fill of the whole output
    vmean_fill_kernel<<<NGRP, 256, 0, stream>>>(v, out);
    // E: WMMA attention for the 40 winners, scattered over the filled output
    topu_attention_kernel<<<NGRP, 384, 0, stream>>>(q, k, v, idx_ws, out);
}